// SmallGNN_11020886081645
// MI455X (gfx1250) — compile-verified
//
#include <hip/hip_runtime.h>
#include <math.h>

typedef __attribute__((ext_vector_type(2))) float v2f;
typedef __attribute__((ext_vector_type(8))) float v8f;

#define NEG_SLOPE 0.2f

// ---------------------------------------------------------------------------
// Generic fill
// ---------------------------------------------------------------------------
__global__ void fill_f32(float* __restrict__ p, float v, int n) {
    int t = blockIdx.x * blockDim.x + threadIdx.x;
    if (t < n) p[t] = v;
}

// ---------------------------------------------------------------------------
// Y[M, Ncols] = X[M, K] @ W[Ncols, K]^T + b   via V_WMMA_F32_16X16X4_F32
// grid.x = M/16 (M % 16 == 0), blockDim.x = (Ncols/16) * 32, 1 wave per N-tile
// ---------------------------------------------------------------------------
__global__ __launch_bounds__(256)
void gemm_wmma_f32(const float* __restrict__ X, const float* __restrict__ W,
                   const float* __restrict__ b, float* __restrict__ Y,
                   int K, int Ncols) {
    const int lane  = threadIdx.x & 31;
    const int wave  = threadIdx.x >> 5;
    const int m0    = blockIdx.x * 16;
    const int n0    = wave * 16;
    if (n0 >= Ncols) return;

    const int row   = m0 + (lane & 15);          // A: M = lane&15
    const int coln  = n0 + (lane & 15);          // B/C/D: N = lane&15
    const int khalf = (lane >> 4) * 2;           // lanes 16-31 hold K=+2,+3

    v8f acc = {};
    for (int k0 = 0; k0 < K; k0 += 4) {
        // A 16x4 f32: 2 consecutive K values per lane (8B aligned: K,k0,khalf even)
        v2f a = *(const v2f*)(X + (size_t)row  * K + k0 + khalf);
        // B 4x16 f32 (W^T): B[k][n] = W[n][k]
        v2f bb = *(const v2f*)(W + (size_t)coln * K + k0 + khalf);
        acc = __builtin_amdgcn_wmma_f32_16x16x4_f32(
                /*neg_a=*/false, a, /*neg_b=*/false, bb,
                /*c_mod=*/(short)0, acc, /*reuse_a=*/false, /*reuse_b=*/false);
    }

    const float bias = b[coln];
    const int rbase = m0 + ((lane >> 4) ? 8 : 0);  // VGPR r: M=r (lanes<16) / M=8+r
#pragma unroll
    for (int r = 0; r < 8; ++r) {
        Y[(size_t)(rbase + r) * Ncols + coln] = acc[r] + bias;
    }
}

// ---------------------------------------------------------------------------
// Layer 1 edge pass A: logits e1[E,4] + segment max (wave per edge, H=4,C=32)
// ---------------------------------------------------------------------------
__global__ __launch_bounds__(256)
void edge_logits1(const int* __restrict__ ei, const float* __restrict__ xl,
                  const float* __restrict__ xr, const float* __restrict__ att,
                  float* __restrict__ e1, float* __restrict__ m1, int E) {
    const int w    = (blockIdx.x * blockDim.x + threadIdx.x) >> 5;
    const int lane = threadIdx.x & 31;
    if (w >= E) return;
    const int src = ei[w];
    const int dst = ei[E + w];

    float v[4];
#pragma unroll
    for (int h = 0; h < 4; ++h) {
        float s = xl[(size_t)src * 128 + h * 32 + lane]
                + xr[(size_t)dst * 128 + h * 32 + lane];
        s = (s > 0.f) ? s : NEG_SLOPE * s;
        v[h] = s * att[h * 32 + lane];
    }
#pragma unroll
    for (int off = 16; off > 0; off >>= 1) {
#pragma unroll
        for (int h = 0; h < 4; ++h) v[h] += __shfl_xor(v[h], off, 32);
    }
    if (lane < 4) {
        e1[(size_t)w * 4 + lane] = v[lane];
        atomicMax(&m1[(size_t)dst * 4 + lane], v[lane]);
    }
}

// ---------------------------------------------------------------------------
// Layer 1 edge pass B: p = exp(e - m[dst]); z[dst] += p (thread per edge*head)
// ---------------------------------------------------------------------------
__global__ __launch_bounds__(256)
void edge_softmax1(const int* __restrict__ ei, float* __restrict__ e1,
                   const float* __restrict__ m1, float* __restrict__ z1, int E) {
    int t = blockIdx.x * blockDim.x + threadIdx.x;
    if (t >= E * 4) return;
    const int e = t >> 2, h = t & 3;
    const int dst = ei[E + e];
    float p = expf(e1[t] - m1[(size_t)dst * 4 + h]);
    e1[t] = p;
    atomicAdd(&z1[(size_t)dst * 4 + h], p);
}

// ---------------------------------------------------------------------------
// Layer 1 edge pass C: out[dst] += alpha * xl[src]  (wave per edge)
// ---------------------------------------------------------------------------
__global__ __launch_bounds__(256)
void edge_aggr1(const int* __restrict__ ei, const float* __restrict__ p1,
                const float* __restrict__ z1, const float* __restrict__ xl,
                float* __restrict__ out, int E) {
    const int w    = (blockIdx.x * blockDim.x + threadIdx.x) >> 5;
    const int lane = threadIdx.x & 31;
    if (w >= E) return;
    const int src = ei[w];
    const int dst = ei[E + w];
#pragma unroll
    for (int h = 0; h < 4; ++h) {
        float alpha = p1[(size_t)w * 4 + h] / (z1[(size_t)dst * 4 + h] + 1e-16f);
        atomicAdd(&out[(size_t)dst * 128 + h * 32 + lane],
                  xl[(size_t)src * 128 + h * 32 + lane] * alpha);
    }
}

// ---------------------------------------------------------------------------
// h = elu(h + bias[c])  in place; cmask = C-1 (C power of two)
// ---------------------------------------------------------------------------
__global__ __launch_bounds__(256)
void elu_bias(float* __restrict__ h, const float* __restrict__ bias,
              int total, int cmask) {
    int t = blockIdx.x * blockDim.x + threadIdx.x;
    if (t >= total) return;
    float v = h[t] + bias[t & cmask];
    h[t] = (v > 0.f) ? v : expm1f(v);
}

// ---------------------------------------------------------------------------
// Layer 2 edge pass A: H=1, C=64 (wave per edge, 2 channels per lane)
// ---------------------------------------------------------------------------
__global__ __launch_bounds__(256)
void edge_logits2(const int* __restrict__ ei, const float* __restrict__ xl,
                  const float* __restrict__ xr, const float* __restrict__ att,
                  float* __restrict__ e2, float* __restrict__ m2, int E) {
    const int w    = (blockIdx.x * blockDim.x + threadIdx.x) >> 5;
    const int lane = threadIdx.x & 31;
    if (w >= E) return;
    const int src = ei[w];
    const int dst = ei[E + w];
    float s0 = xl[(size_t)src * 64 + lane]      + xr[(size_t)dst * 64 + lane];
    float s1 = xl[(size_t)src * 64 + lane + 32] + xr[(size_t)dst * 64 + lane + 32];
    s0 = (s0 > 0.f) ? s0 : NEG_SLOPE * s0;
    s1 = (s1 > 0.f) ? s1 : NEG_SLOPE * s1;
    float v = s0 * att[lane] + s1 * att[lane + 32];
#pragma unroll
    for (int off = 16; off > 0; off >>= 1) v += __shfl_xor(v, off, 32);
    if (lane == 0) {
        e2[w] = v;
        atomicMax(&m2[dst], v);
    }
}

__global__ __launch_bounds__(256)
void edge_softmax2(const int* __restrict__ ei, float* __restrict__ e2,
                   const float* __restrict__ m2, float* __restrict__ z2, int E) {
    int t = blockIdx.x * blockDim.x + threadIdx.x;
    if (t >= E) return;
    const int dst = ei[E + t];
    float p = expf(e2[t] - m2[dst]);
    e2[t] = p;
    atomicAdd(&z2[dst], p);
}

__global__ __launch_bounds__(256)
void edge_aggr2(const int* __restrict__ ei, const float* __restrict__ p2,
                const float* __restrict__ z2, const float* __restrict__ xl,
                float* __restrict__ out, int E) {
    const int w    = (blockIdx.x * blockDim.x + threadIdx.x) >> 5;
    const int lane = threadIdx.x & 31;
    if (w >= E) return;
    const int src = ei[w];
    const int dst = ei[E + w];
    float alpha = p2[w] / (z2[dst] + 1e-16f);
    atomicAdd(&out[(size_t)dst * 64 + lane],
              xl[(size_t)src * 64 + lane] * alpha);
    atomicAdd(&out[(size_t)dst * 64 + lane + 32],
              xl[(size_t)src * 64 + lane + 32] * alpha);
}

__global__ __launch_bounds__(256)
void add_bias_out(const float* __restrict__ acc, const float* __restrict__ bias,
                  float* __restrict__ out, int total, int cmask) {
    int t = blockIdx.x * blockDim.x + threadIdx.x;
    if (t >= total) return;
    out[t] = acc[t] + bias[t & cmask];
}

// ---------------------------------------------------------------------------
// Host-side orchestration
// ---------------------------------------------------------------------------
extern "C" void kernel_launch(void* const* d_in, const int* in_sizes, int n_in,
                              void* d_out, int out_size, void* d_ws, size_t ws_size,
                              hipStream_t stream) {
    const float* x     = (const float*)d_in[0];
    const int*   ei    = (const int*)  d_in[1];
    const float* Wl1   = (const float*)d_in[2];
    const float* bl1   = (const float*)d_in[3];
    const float* Wr1   = (const float*)d_in[4];
    const float* br1   = (const float*)d_in[5];
    const float* att1  = (const float*)d_in[6];
    const float* bias1 = (const float*)d_in[7];
    const float* Wl2   = (const float*)d_in[8];
    const float* bl2   = (const float*)d_in[9];
    const float* Wr2   = (const float*)d_in[10];
    const float* br2   = (const float*)d_in[11];
    const float* att2  = (const float*)d_in[12];
    const float* bias2 = (const float*)d_in[13];

    const int N = in_sizes[0] / 64;   // 50000
    const int E = in_sizes[1] / 2;    // 850000

    float* ws = (float*)d_ws;
    float* xl1  = ws; ws += (size_t)N * 128;
    float* xr1  = ws; ws += (size_t)N * 128;
    float* hbuf = ws; ws += (size_t)N * 128;   // aggregation acc -> elu -> layer2 input
    float* e1   = ws; ws += (size_t)E * 4;
    float* m1   = ws; ws += (size_t)N * 4;
    float* z1   = ws; ws += (size_t)N * 4;
    float* xl2  = ws; ws += (size_t)N * 64;
    float* xr2  = ws; ws += (size_t)N * 64;
    float* e2   = ws; ws += (size_t)E;
    float* m2   = ws; ws += (size_t)N;
    float* z2   = ws; ws += (size_t)N;
    float* out2 = ws; ws += (size_t)N * 64;

    const int TB = 256;
    const int waveBlocks = (E + 7) / 8;   // wave(32) per edge, 8 waves per block

    // --- init accumulators (every call: graph-replay safe) ---
    fill_f32<<<(N * 4   + TB - 1) / TB, TB, 0, stream>>>(m1, -INFINITY, N * 4);
    fill_f32<<<(N * 4   + TB - 1) / TB, TB, 0, stream>>>(z1, 0.f,       N * 4);
    fill_f32<<<(N * 128 + TB - 1) / TB, TB, 0, stream>>>(hbuf, 0.f,     N * 128);
    fill_f32<<<(N       + TB - 1) / TB, TB, 0, stream>>>(m2, -INFINITY, N);
    fill_f32<<<(N       + TB - 1) / TB, TB, 0, stream>>>(z2, 0.f,       N);
    fill_f32<<<(N * 64  + TB - 1) / TB, TB, 0, stream>>>(out2, 0.f,     N * 64);

    // --- layer 1: dense projections (WMMA) ---
    gemm_wmma_f32<<<N / 16, 8 * 32, 0, stream>>>(x, Wl1, bl1, xl1, 64, 128);
    gemm_wmma_f32<<<N / 16, 8 * 32, 0, stream>>>(x, Wr1, br1, xr1, 64, 128);

    // --- layer 1: edge phase ---
    edge_logits1 <<<waveBlocks, TB, 0, stream>>>(ei, xl1, xr1, att1, e1, m1, E);
    edge_softmax1<<<((size_t)E * 4 + TB - 1) / TB, TB, 0, stream>>>(ei, e1, m1, z1, E);
    edge_aggr1   <<<waveBlocks, TB, 0, stream>>>(ei, e1, z1, xl1, hbuf, E);
    elu_bias     <<<((size_t)N * 128 + TB - 1) / TB, TB, 0, stream>>>(hbuf, bias1, N * 128, 127);

    // --- layer 2: dense projections (WMMA) ---
    gemm_wmma_f32<<<N / 16, 4 * 32, 0, stream>>>(hbuf, Wl2, bl2, xl2, 128, 64);
    gemm_wmma_f32<<<N / 16, 4 * 32, 0, stream>>>(hbuf, Wr2, br2, xr2, 128, 64);

    // --- layer 2: edge phase ---
    edge_logits2 <<<waveBlocks, TB, 0, stream>>>(ei, xl2, xr2, att2, e2, m2, E);
    edge_softmax2<<<(E + TB - 1) / TB, TB, 0, stream>>>(ei, e2, m2, z2, E);
    edge_aggr2   <<<waveBlocks, TB, 0, stream>>>(ei, e2, z2, xl2, out2, E);

    // --- final: +bias2 (H=1 so mean over heads is identity) ---
    add_bias_out <<<((size_t)N * 64 + TB - 1) / TB, TB, 0, stream>>>(out2, bias2, (float*)d_out, N * 64, 63);
}